// EdgeConv_33998961115944
// MI455X (gfx1250) — compile-verified
//
#include <hip/hip_runtime.h>
#include <math.h>

typedef __attribute__((ext_vector_type(2))) float v2f;
typedef __attribute__((ext_vector_type(8))) float v8f;

#define B_   8
#define C_   64
#define N_   4096
#define O_   64
#define K_NN 20
#define EPS_ 1e-5f
#define SLOPE_ 0.2f
#define NEG_INF -3.4e38f

static __device__ __forceinline__ v8f wmma_f32(v2f a, v2f b, v8f c) {
    // V_WMMA_F32_16X16X4_F32 : D = A(16x4) * B(4x16) + C(16x16), full fp32
    return __builtin_amdgcn_wmma_f32_16x16x4_f32(false, a, false, b, (short)0, c,
                                                 false, false);
}

// ---------------------------------------------------------------- squared norms
__global__ void __launch_bounds__(256) k_sqnorm(const float* __restrict__ x,
                                                float* __restrict__ xx) {
    int t = blockIdx.x * blockDim.x + threadIdx.x;   // b*N + n
    if (t >= B_ * N_) return;
    int b = t / N_, n = t % N_;
    const float* xb = x + (size_t)b * C_ * N_;
    float s = 0.f;
#pragma unroll
    for (int c = 0; c < C_; ++c) {
        float v = xb[(size_t)c * N_ + n];
        s += v * v;
    }
    xx[t] = s;
}

// ---------------------------------------------------------------- WMMA KNN
// Block = 128 threads = 4 waves; each wave owns 16 rows of the Gram matrix.
// pairwise(n,m) = 2*dot(x_n,x_m) - |x_n|^2 - |x_m|^2 ; keep top-20 per row.
// Scan uses all 32 lanes: lane r handles row r cols [0,8), lane r+16 handles
// row r cols [8,16); the two sorted top-20 lists are merged once at the end.
__global__ void __launch_bounds__(128) k_knn(const float* __restrict__ x,
                                             const float* __restrict__ xx,
                                             int* __restrict__ idx) {
    const int lane = threadIdx.x & 31;
    const int wave = threadIdx.x >> 5;
    const int b    = blockIdx.x / (N_ / 64);
    const int rb   = (blockIdx.x % (N_ / 64)) * 64;
    const int row0 = rb + wave * 16;

    const float* xb  = x  + (size_t)b * C_ * N_;
    const float* xxb = xx + (size_t)b * N_;

    const int hs  = lane >> 4;      // half-wave select (0/1)
    const int l16 = lane & 15;
    const int jb  = hs * 8;         // this lane's column sub-range in the tile

    // Preload A tile (16 rows x 64 channels) in WMMA A-layout:
    // lanes 0-15: M=l16, K = 4s+0,4s+1 ; lanes 16-31: M=l16, K = 4s+2,4s+3
    v2f A[16];
#pragma unroll
    for (int s = 0; s < 16; ++s) {
        int c0 = 4 * s + hs * 2;
        A[s].x = xb[(size_t)(c0 + 0) * N_ + row0 + l16];
        A[s].y = xb[(size_t)(c0 + 1) * N_ + row0 + l16];
    }

    float xr[8];                    // row norms for this lane's accumulator rows
#pragma unroll
    for (int v = 0; v < 8; ++v) xr[v] = xxb[row0 + v + hs * 8];

    float vals[K_NN];
    int   ids[K_NN];
#pragma unroll
    for (int t = 0; t < K_NN; ++t) { vals[t] = NEG_INF; ids[t] = 0; }

    __shared__ float tile[2][4][16][17];    // double-buffered distance tiles

    int p = 0;
    for (int m0 = 0; m0 < N_; m0 += 16, p ^= 1) {
        v2f Bv[16];
#pragma unroll
        for (int s = 0; s < 16; ++s) {
            int c0 = 4 * s + hs * 2;
            Bv[s].x = xb[(size_t)(c0 + 0) * N_ + m0 + l16];
            Bv[s].y = xb[(size_t)(c0 + 1) * N_ + m0 + l16];
        }
        v8f acc = {};
#pragma unroll
        for (int s = 0; s < 16; ++s) acc = wmma_f32(A[s], Bv[s], acc);

        float xc = xxb[m0 + l16];
#pragma unroll
        for (int v = 0; v < 8; ++v) {
            int r = v + hs * 8;
            tile[p][wave][r][l16] = 2.0f * acc[v] - xr[v] - xc;
        }
        __syncthreads();    // tile[p] complete; also proves tile[p^1] scans done
        for (int j = 0; j < 8; ++j) {
            float d = tile[p][wave][l16][jb + j];
            if (d > vals[K_NN - 1]) {       // candidate beats current 20th
                float cv = d; int ci = m0 + jb + j;
#pragma unroll
                for (int t = 0; t < K_NN; ++t) {
                    if (cv > vals[t]) {     // strict > : lowest-index wins ties
                        float tv = vals[t]; int ti = ids[t];
                        vals[t] = cv; ids[t] = ci;
                        cv = tv; ci = ti;
                    }
                }
            }
        }
    }

    // Merge each row's two sorted top-20 lists (cols 0-7 stream vs 8-15 stream).
    __shared__ float mv[128][K_NN];
    __shared__ int   mi[128][K_NN];
#pragma unroll
    for (int t = 0; t < K_NN; ++t) {
        mv[threadIdx.x][t] = vals[t];
        mi[threadIdx.x][t] = ids[t];
    }
    __syncthreads();
    if (lane < 16) {
        const float* av = mv[wave * 32 + lane];
        const int*   ai = mi[wave * 32 + lane];
        const float* bv = mv[wave * 32 + lane + 16];
        const int*   bi = mi[wave * 32 + lane + 16];
        int* op = idx + ((size_t)b * N_ + row0 + lane) * K_NN;
        int ia = 0, ib2 = 0;
        for (int t = 0; t < K_NN; ++t) {
            float fa = av[ia], fb = bv[ib2];
            bool takeA = (fa > fb) || (fa == fb && ai[ia] < bi[ib2]);
            op[t] = takeA ? ai[ia++] : bi[ib2++];
        }
    }
}

// ---------------------------------------------------------------- combined weights
// Wc[o][c]      = W[o][c]              (o < 64)   -> multiplies neighbor
// Wc[o+64][c]   = W[o][64+c] - W[o][c]            -> multiplies center
__global__ void __launch_bounds__(256) k_wc(const float* __restrict__ W,
                                            float* __restrict__ Wc) {
    int t = blockIdx.x * blockDim.x + threadIdx.x;
    if (t >= 128 * 64) return;
    int op = t / 64, c = t % 64;
    float v;
    if (op < 64) v = W[(size_t)op * 128 + c];
    else         v = W[(size_t)(op - 64) * 128 + 64 + c] -
                     W[(size_t)(op - 64) * 128 + c];
    Wc[t] = v;
}

// ---------------------------------------------------------------- WMMA projection
// P[b][o'][n] = sum_c Wc[o'][c] * x[b][c][n] ; o' in [0,128)
// Block = 256 threads = 8 waves; wave w computes o-tile w, block covers one n-tile.
__global__ void __launch_bounds__(256) k_proj(const float* __restrict__ x,
                                              const float* __restrict__ Wc,
                                              float* __restrict__ P) {
    const int lane = threadIdx.x & 31;
    const int wave = threadIdx.x >> 5;
    const int b    = blockIdx.x / (N_ / 16);
    const int n0   = (blockIdx.x % (N_ / 16)) * 16;
    const int o0   = wave * 16;
    const float* xb = x + (size_t)b * C_ * N_;
    const int hs = lane >> 4, l16 = lane & 15;

    v8f acc = {};
#pragma unroll
    for (int s = 0; s < 16; ++s) {
        int c0 = 4 * s + hs * 2;
        v2f a, bb;
        a.x  = Wc[(size_t)(o0 + l16) * 64 + c0 + 0];
        a.y  = Wc[(size_t)(o0 + l16) * 64 + c0 + 1];
        bb.x = xb[(size_t)(c0 + 0) * N_ + n0 + l16];
        bb.y = xb[(size_t)(c0 + 1) * N_ + n0 + l16];
        acc = wmma_f32(a, bb, acc);
    }
    float* Pb = P + (size_t)b * 128 * N_;
#pragma unroll
    for (int v = 0; v < 8; ++v) {
        int m = o0 + v + hs * 8;
        Pb[(size_t)m * N_ + n0 + l16] = acc[v];
    }
}

// ---------------------------------------------------------------- BN stat buffers
__global__ void k_zero(float* __restrict__ S) {
    if (threadIdx.x < 128) S[threadIdx.x] = 0.f;
}

// S[o] = sum y, S[64+o] = sum y^2 over (b,n,k);  y = P1[idx] + P2[n]
__global__ void __launch_bounds__(256) k_stats(const float* __restrict__ P,
                                               const int* __restrict__ idx,
                                               float* __restrict__ S) {
    const int chunks = N_ / 256;
    const int bo = blockIdx.x / chunks;
    const int n  = (blockIdx.x % chunks) * 256 + threadIdx.x;
    const int b = bo / O_, o = bo % O_;

    const float* P1 = P + (size_t)b * 128 * N_ + (size_t)o * N_;
    const float  p2 = P1[(size_t)64 * N_ + n];
    const int*   ip = idx + ((size_t)b * N_ + n) * K_NN;

    float s1 = 0.f, s2 = 0.f;
#pragma unroll
    for (int k = 0; k < K_NN; ++k) {
        float y = P1[ip[k]] + p2;
        s1 += y; s2 += y * y;
    }
    __shared__ float r1[256], r2[256];
    r1[threadIdx.x] = s1; r2[threadIdx.x] = s2;
    __syncthreads();
    for (int st = 128; st > 0; st >>= 1) {
        if (threadIdx.x < st) {
            r1[threadIdx.x] += r1[threadIdx.x + st];
            r2[threadIdx.x] += r2[threadIdx.x + st];
        }
        __syncthreads();
    }
    if (threadIdx.x == 0) {
        atomicAdd(&S[o],      r1[0]);
        atomicAdd(&S[64 + o], r2[0]);
    }
}

// ---------------------------------------------------------------- finalize
__global__ void __launch_bounds__(256) k_final(const float* __restrict__ P,
                                               const int* __restrict__ idx,
                                               const float* __restrict__ S,
                                               const float* __restrict__ gamma,
                                               const float* __restrict__ beta,
                                               float* __restrict__ out) {
    const int chunks = N_ / 256;
    const int bo = blockIdx.x / chunks;
    const int n  = (blockIdx.x % chunks) * 256 + threadIdx.x;
    const int b = bo / O_, o = bo % O_;

    const float cnt  = (float)((size_t)B_ * N_ * K_NN);
    const float mean = S[o] / cnt;
    const float var  = S[64 + o] / cnt - mean * mean;
    const float g    = gamma[o] * rsqrtf(var + EPS_);
    const float be   = beta[o];

    const float* P1 = P + (size_t)b * 128 * N_ + (size_t)o * N_;
    const float  p2 = P1[(size_t)64 * N_ + n];
    const int*   ip = idx + ((size_t)b * N_ + n) * K_NN;

    float best = NEG_INF;
#pragma unroll
    for (int k = 0; k < K_NN; ++k) {
        float y = P1[ip[k]] + p2;
        float z = (y - mean) * g + be;
        z = (z >= 0.f) ? z : SLOPE_ * z;
        best = fmaxf(best, z);
    }
    out[((size_t)b * O_ + o) * N_ + n] = best;
}

// ---------------------------------------------------------------- launcher
extern "C" void kernel_launch(void* const* d_in, const int* in_sizes, int n_in,
                              void* d_out, int out_size, void* d_ws, size_t ws_size,
                              hipStream_t stream) {
    const float* x     = (const float*)d_in[0];   // (8,64,4096)
    const float* W     = (const float*)d_in[1];   // (64,128)
    const float* gamma = (const float*)d_in[2];   // (64,)
    const float* beta  = (const float*)d_in[3];   // (64,)
    float* out = (float*)d_out;                   // (8,64,4096)

    char* ws = (char*)d_ws;
    size_t off = 0;
    float* xx = (float*)(ws + off); off += (size_t)B_ * N_ * sizeof(float);        // 128 KB
    int*   id = (int*)  (ws + off); off += (size_t)B_ * N_ * K_NN * sizeof(int);   // 2.5 MB
    float* Wc = (float*)(ws + off); off += (size_t)128 * 64 * sizeof(float);       // 32 KB
    float* P  = (float*)(ws + off); off += (size_t)B_ * 128 * N_ * sizeof(float);  // 16 MB
    float* S  = (float*)(ws + off); off += 128 * sizeof(float);
    (void)ws_size; (void)in_sizes; (void)n_in; (void)out_size;

    k_sqnorm<<<(B_ * N_ + 255) / 256, 256, 0, stream>>>(x, xx);
    k_knn  <<<B_ * (N_ / 64), 128, 0, stream>>>(x, xx, id);
    k_wc   <<<(128 * 64 + 255) / 256, 256, 0, stream>>>(W, Wc);
    k_proj <<<B_ * (N_ / 16), 256, 0, stream>>>(x, Wc, P);
    k_zero <<<1, 128, 0, stream>>>(S);
    k_stats<<<B_ * O_ * (N_ / 256), 256, 0, stream>>>(P, id, S);
    k_final<<<B_ * O_ * (N_ / 256), 256, 0, stream>>>(P, id, S, gamma, beta, out);
}